// GINEConv__61237643706859
// MI455X (gfx1250) — compile-verified
//
#include <hip/hip_runtime.h>
#include <math.h>

typedef __attribute__((ext_vector_type(2))) float v2f;
typedef __attribute__((ext_vector_type(8))) float v8f;

#define N_NODES 200000
#define N_EDGES 6400000
#define ND 16
#define ED 8
#define NG 256
#define HD 5
#define EPS_GN 1e-5f

// ---------------------------------------------------------------------------
// Workspace layout (floats):
//   agg1 [N*16] | agg2 [N*5] | gsum [G*5] | gssq [G*5] | gcnt [G]
//   h [N*5] | sub [N*5] | h2 [N*5]
// The prefix up through gcnt must be zeroed each launch.
// ---------------------------------------------------------------------------

__global__ __launch_bounds__(256) void zero_kernel(float4* __restrict__ p, int n4) {
    int i = blockIdx.x * blockDim.x + threadIdx.x;
    if (i < n4) p[i] = make_float4(0.f, 0.f, 0.f, 0.f);
}

// ---------------------------------------------------------------------------
// GINE edge pass 1: per wave, 16 edges.
//   D[16 edges][16 feats] = A(16x8 edge_attr) @ B(8x16 = W.T) + bias
// done as two chained V_WMMA_F32_16X16X4_F32 (K=0..3, K=4..7), exact fp32.
// Then m = relu(x[src] + D) and atomic scatter-add into agg1[dst].
// A layout (ISA 7.12.2, 32-bit A 16x4): lane<16 -> M=lane, K={0,1};
// lane>=16 -> M=lane-16, K={2,3}.  B mirrors with N=lane&15.
// D layout: VGPR r -> row M=r (lanes<16) / M=r+8 (lanes>=16), lane -> col N.
// ---------------------------------------------------------------------------
__global__ __launch_bounds__(256) void gine_edge1(
    const float* __restrict__ eattr, const int* __restrict__ ei,
    const float* __restrict__ x, const float* __restrict__ We,
    const float* __restrict__ be, float* __restrict__ agg)
{
    const int lane = threadIdx.x & 31;
    const int wave = blockIdx.x * (blockDim.x >> 5) + (threadIdx.x >> 5);
    const int ebase = wave << 4;
    if (ebase >= N_EDGES) return;             // wave-uniform: EXEC stays full

    const int r16 = lane & 15;
    const int k0  = (lane >= 16) ? 2 : 0;
    const int e   = ebase + r16;

    v2f a0 = *(const v2f*)(eattr + e * ED + k0);
    v2f a1 = *(const v2f*)(eattr + e * ED + k0 + 4);
    v2f b0 = *(const v2f*)(We + r16 * ED + k0);
    v2f b1 = *(const v2f*)(We + r16 * ED + k0 + 4);

    float bias = be[r16];
    v8f c;
#pragma unroll
    for (int r = 0; r < 8; r++) c[r] = bias;

    c = __builtin_amdgcn_wmma_f32_16x16x4_f32(false, a0, false, b0, (short)0, c, false, false);
    c = __builtin_amdgcn_wmma_f32_16x16x4_f32(false, a1, false, b1, (short)0, c, false, false);

    const int* __restrict__ srcp = ei;
    const int* __restrict__ dstp = ei + N_EDGES;
    const int mb = (lane >= 16) ? 8 : 0;
#pragma unroll
    for (int r = 0; r < 8; r++) {
        int ee = ebase + mb + r;
        int s = srcp[ee];
        int d = dstp[ee];
        float v = c[r] + x[s * ND + r16];       // x gather: L2-resident (12.8MB)
        v = fmaxf(v, 0.0f);
        atomicAdd(agg + d * ND + r16, v);       // scatter into L2-resident agg1
    }
}

// ---------------------------------------------------------------------------
// GINE edge pass 2: same tiling, but only HD=5 output columns are real.
// Clamp B loads for n>=5 (column independence makes those D columns junk we
// simply never scatter); divergent tail happens AFTER the WMMAs.
// ---------------------------------------------------------------------------
__global__ __launch_bounds__(256) void gine_edge2(
    const float* __restrict__ eattr, const int* __restrict__ ei,
    const float* __restrict__ hsrc, const float* __restrict__ We,
    const float* __restrict__ be, float* __restrict__ agg)
{
    const int lane = threadIdx.x & 31;
    const int wave = blockIdx.x * (blockDim.x >> 5) + (threadIdx.x >> 5);
    const int ebase = wave << 4;
    if (ebase >= N_EDGES) return;

    const int r16 = lane & 15;
    const int k0  = (lane >= 16) ? 2 : 0;
    const int e   = ebase + r16;
    const int nc  = (r16 < HD) ? r16 : 0;       // clamp to stay in-bounds of W[5][8]

    v2f a0 = *(const v2f*)(eattr + e * ED + k0);
    v2f a1 = *(const v2f*)(eattr + e * ED + k0 + 4);
    v2f b0 = *(const v2f*)(We + nc * ED + k0);
    v2f b1 = *(const v2f*)(We + nc * ED + k0 + 4);

    float bias = be[nc];
    v8f c;
#pragma unroll
    for (int r = 0; r < 8; r++) c[r] = bias;

    c = __builtin_amdgcn_wmma_f32_16x16x4_f32(false, a0, false, b0, (short)0, c, false, false);
    c = __builtin_amdgcn_wmma_f32_16x16x4_f32(false, a1, false, b1, (short)0, c, false, false);

    if (r16 < HD) {
        const int* __restrict__ srcp = ei;
        const int* __restrict__ dstp = ei + N_EDGES;
        const int mb = (lane >= 16) ? 8 : 0;
#pragma unroll
        for (int r = 0; r < 8; r++) {
            int ee = ebase + mb + r;
            int s = srcp[ee];
            int d = dstp[ee];
            float v = c[r] + hsrc[s * HD + r16];
            v = fmaxf(v, 0.0f);
            atomicAdd(agg + d * HD + r16, v);
        }
    }
}

// ---------------------------------------------------------------------------
// Node pass 1: h = ((1+eps1)*x + agg1) @ nn1_W.T + nn1_b  -> h[N*5]
// Also per-graph sums + counts. batch_idx is sorted so (g - g_of_first_node)
// fits in [0,255]: block-local LDS table with ds_add_f32, then a sparse flush.
// ---------------------------------------------------------------------------
__global__ __launch_bounds__(256) void node_pass1(
    const float* __restrict__ x, const float* __restrict__ agg1,
    const int* __restrict__ bidx, const float* __restrict__ eps1p,
    const float* __restrict__ Wn, const float* __restrict__ bn,
    float* __restrict__ h, float* __restrict__ gsum, float* __restrict__ gcnt)
{
    __shared__ float tbl[256][6];
    const int t = threadIdx.x;
    const int i = blockIdx.x * 256 + t;
#pragma unroll
    for (int j = 0; j < 6; j++) tbl[t][j] = 0.0f;
    __syncthreads();

    const int gmin = bidx[blockIdx.x * 256];    // first node of block exists
    if (i < N_NODES) {
        const float f = 1.0f + eps1p[0];
        float hp[ND];
#pragma unroll
        for (int k = 0; k < ND; k++) hp[k] = f * x[i * ND + k] + agg1[i * ND + k];
        float hv[HD];
#pragma unroll
        for (int j = 0; j < HD; j++) {
            float acc = bn[j];
#pragma unroll
            for (int k = 0; k < ND; k++) acc += Wn[j * ND + k] * hp[k];
            hv[j] = acc;
            h[i * HD + j] = acc;
        }
        const int gl = bidx[i] - gmin;          // in [0,255] (sorted, g<=255)
#pragma unroll
        for (int j = 0; j < HD; j++) atomicAdd(&tbl[gl][j], hv[j]);
        atomicAdd(&tbl[gl][5], 1.0f);
    }
    __syncthreads();

    const float cn = tbl[t][5];
    if (cn != 0.0f) {
        const int g = gmin + t;
#pragma unroll
        for (int j = 0; j < HD; j++) atomicAdd(gsum + g * HD + j, tbl[t][j]);
        atomicAdd(gcnt + g, cn);
    }
}

__global__ void gmean_fin(float* __restrict__ gsum, const float* __restrict__ gcnt) {
    const int g = threadIdx.x;                  // <<<1,256>>>
    const float c = fmaxf(gcnt[g], 1.0f);
#pragma unroll
    for (int j = 0; j < HD; j++) gsum[g * HD + j] /= c;
}

// Node pass 2: sub = h - mean_scale*mean[g]; accumulate sum(sub^2) per graph.
__global__ __launch_bounds__(256) void node_pass2(
    const float* __restrict__ h, const int* __restrict__ bidx,
    const float* __restrict__ gmean, const float* __restrict__ ms,
    float* __restrict__ sub, float* __restrict__ gssq)
{
    __shared__ float tbl[256][6];
    const int t = threadIdx.x;
    const int i = blockIdx.x * 256 + t;
#pragma unroll
    for (int j = 0; j < 6; j++) tbl[t][j] = 0.0f;
    __syncthreads();

    const int gmin = bidx[blockIdx.x * 256];
    if (i < N_NODES) {
        const int g = bidx[i];
        const int gl = g - gmin;
#pragma unroll
        for (int j = 0; j < HD; j++) {
            float s = h[i * HD + j] - ms[j] * gmean[g * HD + j];
            sub[i * HD + j] = s;
            atomicAdd(&tbl[gl][j], s * s);
        }
        tbl[gl][5] = 1.0f;                      // touched flag (benign race)
    }
    __syncthreads();

    if (tbl[t][5] != 0.0f) {
        const int g = gmin + t;
#pragma unroll
        for (int j = 0; j < HD; j++) atomicAdd(gssq + g * HD + j, tbl[t][j]);
    }
}

__global__ void gvar_fin(float* __restrict__ gssq, const float* __restrict__ gcnt) {
    const int g = threadIdx.x;                  // <<<1,256>>>
    const float c = fmaxf(gcnt[g], 1.0f);
#pragma unroll
    for (int j = 0; j < HD; j++) {
        float v = gssq[g * HD + j] / c;
        gssq[g * HD + j] = rsqrtf(v + EPS_GN);  // store rstd in place
    }
}

// Node pass 3: h2 = relu(weight*sub*rstd[g] + bias)
__global__ __launch_bounds__(256) void node_pass3(
    const float* __restrict__ sub, const int* __restrict__ bidx,
    const float* __restrict__ rstd, const float* __restrict__ w,
    const float* __restrict__ b, float* __restrict__ h2)
{
    const int idx = blockIdx.x * blockDim.x + threadIdx.x;
    if (idx >= N_NODES * HD) return;
    const int i = idx / HD;
    const int j = idx - i * HD;
    const int g = bidx[i];
    float v = w[j] * sub[idx] * rstd[g * HD + j] + b[j];
    h2[idx] = fmaxf(v, 0.0f);
}

// Final: out = sigmoid(((1+eps2)*h2 + agg2) @ nn2_W.T + nn2_b)
__global__ __launch_bounds__(256) void node_pass4(
    const float* __restrict__ h2, const float* __restrict__ agg2,
    const float* __restrict__ eps2p, const float* __restrict__ Wn,
    const float* __restrict__ bn, float* __restrict__ out)
{
    const int i = blockIdx.x * blockDim.x + threadIdx.x;
    if (i >= N_NODES) return;
    const float f = 1.0f + eps2p[0];
    float acc = bn[0];
#pragma unroll
    for (int j = 0; j < HD; j++)
        acc += Wn[j] * (f * h2[i * HD + j] + agg2[i * HD + j]);
    out[i] = 1.0f / (1.0f + expf(-acc));
}

extern "C" void kernel_launch(void* const* d_in, const int* in_sizes, int n_in,
                              void* d_out, int out_size, void* d_ws, size_t ws_size,
                              hipStream_t stream) {
    const float* x     = (const float*)d_in[0];
    const int*   ei    = (const int*)  d_in[1];   // [2*E], src then dst (int32)
    const float* eattr = (const float*)d_in[2];
    const int*   bidx  = (const int*)  d_in[3];
    const float* eps1  = (const float*)d_in[4];
    const float* W1e   = (const float*)d_in[5];
    const float* b1e   = (const float*)d_in[6];
    const float* W1n   = (const float*)d_in[7];
    const float* b1n   = (const float*)d_in[8];
    const float* gnw   = (const float*)d_in[9];
    const float* gnb   = (const float*)d_in[10];
    const float* gnms  = (const float*)d_in[11];
    const float* eps2  = (const float*)d_in[12];
    const float* W2e   = (const float*)d_in[13];
    const float* b2e   = (const float*)d_in[14];
    const float* W2n   = (const float*)d_in[15];
    const float* b2n   = (const float*)d_in[16];
    float* out = (float*)d_out;

    float* ws   = (float*)d_ws;
    float* agg1 = ws;                                  // N*16
    float* agg2 = agg1 + (size_t)N_NODES * ND;         // N*5
    float* gsum = agg2 + (size_t)N_NODES * HD;         // G*5
    float* gssq = gsum + NG * HD;                      // G*5
    float* gcnt = gssq + NG * HD;                      // G
    float* h    = gcnt + NG;                           // N*5
    float* sub  = h   + (size_t)N_NODES * HD;          // N*5
    float* h2   = sub + (size_t)N_NODES * HD;          // N*5

    const size_t zero_floats = (size_t)N_NODES * ND + (size_t)N_NODES * HD
                             + 2 * NG * HD + NG;       // = 4,202,816 (div by 4)
    const int n4 = (int)(zero_floats / 4);
    zero_kernel<<<(n4 + 255) / 256, 256, 0, stream>>>((float4*)ws, n4);

    const int edge_waves  = N_EDGES / 16;              // 400,000 (exact)
    const int edge_blocks = (edge_waves + 7) / 8;      // 8 waves / 256-thread block
    const int node_blocks = (N_NODES + 255) / 256;
    const int nb5         = (N_NODES * HD + 255) / 256;

    gine_edge1<<<edge_blocks, 256, 0, stream>>>(eattr, ei, x, W1e, b1e, agg1);
    node_pass1<<<node_blocks, 256, 0, stream>>>(x, agg1, bidx, eps1, W1n, b1n,
                                                h, gsum, gcnt);
    gmean_fin<<<1, 256, 0, stream>>>(gsum, gcnt);
    node_pass2<<<node_blocks, 256, 0, stream>>>(h, bidx, gsum, gnms, sub, gssq);
    gvar_fin<<<1, 256, 0, stream>>>(gssq, gcnt);
    node_pass3<<<nb5, 256, 0, stream>>>(sub, bidx, gssq, gnw, gnb, h2);
    gine_edge2<<<edge_blocks, 256, 0, stream>>>(eattr, ei, h2, W2e, b2e, agg2);
    node_pass4<<<node_blocks, 256, 0, stream>>>(h2, agg2, eps2, W2n, b2n, out);
}